// NeuralAttention_63428077027729
// MI455X (gfx1250) — compile-verified
//
#include <hip/hip_runtime.h>

typedef __attribute__((ext_vector_type(2))) float v2f;
typedef __attribute__((ext_vector_type(4))) float v4f;
typedef __attribute__((ext_vector_type(8))) float v8f;

#define HH 2
#define SS 512
#define DKK 64
#define DHH 256

// ---------------------------------------------------------------------------
// Kernel 1: projections via V_WMMA_F32_16X16X4_F32 (fp32 WMMA keeps reference
// precision; K=64 -> 16 chained WMMAs per 16x16 output tile).
//   qp = Q @ W1[:64]  + b1      kp = K @ W1[64:]
// One wave (32 lanes) per 16x16 tile.  EXEC all-1s, no divergence.
// ---------------------------------------------------------------------------
__global__ __launch_bounds__(32)
void proj_wmma_f32(const float* __restrict__ Qm, const float* __restrict__ Km,
                   const float* __restrict__ W1, const float* __restrict__ b1,
                   float* __restrict__ qp, float* __restrict__ kp)
{
    const int lane = threadIdx.x;     // 0..31
    const int mn   = lane & 15;       // A-row m / B-col n / C-col n
    const int kh   = lane >> 4;       // K-half select (ISA 16x4 f32 A layout)

    int t = blockIdx.x;
    const int tilesPerMat = HH * (SS / 16) * (DHH / 16);   // 1024
    const int which = (t >= tilesPerMat) ? 1 : 0;          // 0 = qp, 1 = kp
    t -= which * tilesPerMat;
    const int h  = t / ((SS / 16) * (DHH / 16));
    const int st = (t / (DHH / 16)) % (SS / 16);
    const int ft = t % (DHH / 16);

    const float* src = which ? Km : Qm;
    const float* w   = W1 + which * (DKK * DHH);           // Wq rows [0,64), Wk rows [64,128)
    float*       dst = which ? kp : qp;

    const float* arow = src + ((h * SS) + st * 16 + mn) * DKK;  // A row m
    const float* wcol = w + (ft * 16 + mn);                     // B column n

    v8f c = {};
    #pragma unroll
    for (int k0 = 0; k0 < DKK; k0 += 4) {
        const int ka = k0 + 2 * kh;
        v2f a, b;
        // A 16x4: vgpr0 = K=k0+2*kh, vgpr1 = K=k0+1+2*kh (lane m = row)
        a.x = arow[ka];
        a.y = arow[ka + 1];
        // B 4x16: same K striping, lane n = column
        b.x = wcol[(size_t)(ka)     * DHH];
        b.y = wcol[(size_t)(ka + 1) * DHH];
        c = __builtin_amdgcn_wmma_f32_16x16x4_f32(false, a, false, b,
                                                  (short)0, c, false, false);
    }

    const float bias = which ? 0.0f : b1[ft * 16 + mn];
    float* drow = dst + (size_t)(h * SS + st * 16) * DHH + ft * 16 + mn;
    #pragma unroll
    for (int r = 0; r < 8; ++r) {
        const int row = r + 8 * kh;          // C/D layout: vgpr r, lanes16-31 -> M=r+8
        drow[(size_t)row * DHH] = c[r] + bias;
    }
}

// ---------------------------------------------------------------------------
// Kernel 2: scores[h,i,j] = sum_f W2[f]*relu(qp[i,f]+kp[j,f]) + b2.
// ReLU inside the reduction => no matmul structure; LDS-tiled VALU kernel.
// 32x32 output tile per block: qp/kp rows staged once, reused 32x.
// Rows padded to 260 floats (1040 B, 16B-aligned) for b128 LDS reads.
// ---------------------------------------------------------------------------
#define TI 32
#define TJ 32
#define RP (DHH + 4)

__global__ __launch_bounds__(256)
void pair_score(const float* __restrict__ qp, const float* __restrict__ kp,
                const float* __restrict__ W2, const float* __restrict__ b2,
                float* __restrict__ out)
{
    __shared__ __align__(16) float qs[TI][RP];
    __shared__ __align__(16) float ks[TJ][RP];
    __shared__ __align__(16) float w2s[DHH];

    const int tid = threadIdx.x;
    int bid = blockIdx.x;
    const int jt = bid % (SS / TJ); bid /= (SS / TJ);
    const int it = bid % (SS / TI); bid /= (SS / TI);
    const int h  = bid;

    const float* qbase = qp + (size_t)(h * SS + it * TI) * DHH;
    const float* kbase = kp + (size_t)(h * SS + jt * TJ) * DHH;

    // Cooperative stage: 32 rows x 64 float4 per array (coalesced b128 loads).
    for (int idx = tid; idx < TI * (DHH / 4); idx += 256) {
        const int row = idx >> 6;            // DHH/4 == 64
        const int c4  = (idx & 63) * 4;
        *(v4f*)&qs[row][c4] = *(const v4f*)(qbase + (size_t)row * DHH + c4);
        *(v4f*)&ks[row][c4] = *(const v4f*)(kbase + (size_t)row * DHH + c4);
    }
    if (tid < DHH) w2s[tid] = W2[tid];
    __syncthreads();

    const int jj  = tid & 31;    // wave covers 32 j's (k-row per lane)
    const int ii0 = tid >> 5;    // 8 base i's; each thread does i = ii0+{0,8,16,24}
    float acc0 = 0.f, acc1 = 0.f, acc2 = 0.f, acc3 = 0.f;

    for (int f = 0; f < DHH; f += 4) {
        const v4f kv = *(const v4f*)&ks[jj][f];
        const v4f wv = *(const v4f*)&w2s[f];
        const v4f q0 = *(const v4f*)&qs[ii0      ][f];
        const v4f q1 = *(const v4f*)&qs[ii0 +  8 ][f];
        const v4f q2 = *(const v4f*)&qs[ii0 + 16 ][f];
        const v4f q3 = *(const v4f*)&qs[ii0 + 24 ][f];
        acc0 += wv.x * fmaxf(q0.x + kv.x, 0.f) + wv.y * fmaxf(q0.y + kv.y, 0.f)
              + wv.z * fmaxf(q0.z + kv.z, 0.f) + wv.w * fmaxf(q0.w + kv.w, 0.f);
        acc1 += wv.x * fmaxf(q1.x + kv.x, 0.f) + wv.y * fmaxf(q1.y + kv.y, 0.f)
              + wv.z * fmaxf(q1.z + kv.z, 0.f) + wv.w * fmaxf(q1.w + kv.w, 0.f);
        acc2 += wv.x * fmaxf(q2.x + kv.x, 0.f) + wv.y * fmaxf(q2.y + kv.y, 0.f)
              + wv.z * fmaxf(q2.z + kv.z, 0.f) + wv.w * fmaxf(q2.w + kv.w, 0.f);
        acc3 += wv.x * fmaxf(q3.x + kv.x, 0.f) + wv.y * fmaxf(q3.y + kv.y, 0.f)
              + wv.z * fmaxf(q3.z + kv.z, 0.f) + wv.w * fmaxf(q3.w + kv.w, 0.f);
    }

    const float bb = b2[0];
    float* orow = out + (size_t)(h * SS + it * TI) * SS + jt * TJ + jj;
    orow[(size_t)(ii0      ) * SS] = acc0 + bb;
    orow[(size_t)(ii0 +  8 ) * SS] = acc1 + bb;
    orow[(size_t)(ii0 + 16 ) * SS] = acc2 + bb;
    orow[(size_t)(ii0 + 24 ) * SS] = acc3 + bb;
}

// ---------------------------------------------------------------------------
extern "C" void kernel_launch(void* const* d_in, const int* in_sizes, int n_in,
                              void* d_out, int out_size, void* d_ws, size_t ws_size,
                              hipStream_t stream)
{
    (void)in_sizes; (void)n_in; (void)out_size; (void)ws_size;
    const float* Qm = (const float*)d_in[0];
    const float* Km = (const float*)d_in[1];
    const float* W1 = (const float*)d_in[2];
    const float* b1 = (const float*)d_in[3];
    const float* W2 = (const float*)d_in[4];
    const float* b2 = (const float*)d_in[5];
    float* out = (float*)d_out;

    float* qp = (float*)d_ws;                         // H*S*DH = 262144 floats (1 MB)
    float* kp = qp + (size_t)HH * SS * DHH;           // 1 MB more

    // 2 matrices * H * (S/16) * (DH/16) = 2048 tiles, one wave each.
    proj_wmma_f32<<<2048, 32, 0, stream>>>(Qm, Km, W1, b1, qp, kp);

    // H * (S/32) * (S/32) = 512 blocks of 256 threads.
    pair_score<<<HH * (SS / TI) * (SS / TJ), 256, 0, stream>>>(qp, kp, W2, b2, out);
}